// EdgeConv_11630771437589
// MI455X (gfx1250) — compile-verified
//
#include <hip/hip_runtime.h>

#define KNN   20
#define ALPHA 0.2f
#define BATCH 8
#define CH    64
#define NPTS  2048
#define OUTC  64
#define FAP   (CH + 4)   // padded fa row: 272B stride (16B aligned, bank-spread)

typedef __attribute__((ext_vector_type(2))) float v2f;
typedef __attribute__((ext_vector_type(8))) float v8f;

// fp32 tensor core op: D(16x16) += A(16x4) * B(4x16)
// A layout: lane l holds M = l&15; VGPR v holds K = 2*(l>>4) + v
// B layout: lane l holds N = l&15; VGPR v holds K = 2*(l>>4) + v
// C/D     : lane l, VGPR r -> (M = r + 8*(l>>4), N = l&15)
__device__ __forceinline__ v8f wmma_f32(v2f a, v2f b, v8f c) {
  return __builtin_amdgcn_wmma_f32_16x16x4_f32(false, a, false, b, (short)0, c,
                                               false, false);
}

// ---------------------------------------------------------------------------
// Kernel 1: transpose x (B,C,N) -> xt (B,N,C) and squared norms xx (B,N)
// ---------------------------------------------------------------------------
__global__ __launch_bounds__(256) void prep_kernel(const float* __restrict__ x,
                                                   float* __restrict__ xt,
                                                   float* __restrict__ xx) {
  int n = blockIdx.x * blockDim.x + threadIdx.x;   // 0 .. B*N-1
  if (n >= BATCH * NPTS) return;
  int b = n / NPTS;
  int p = n - b * NPTS;
  const float* xb = x + (size_t)b * CH * NPTS;
  float s = 0.f;
#pragma unroll
  for (int c = 0; c < CH; ++c) {
    float v = xb[c * NPTS + p];
    xt[(size_t)n * CH + c] = v;
    s += v * v;
  }
  xx[n] = s;
}

// ---------------------------------------------------------------------------
// Kernel 2: fused pairwise-distance (fp32 WMMA) + streaming top-20 per row.
// One wave handles 16 query rows; A fragments hoisted to registers (loop
// invariant), B fragments double-buffered so global loads overlap WMMA +
// selection. neg_dist[m][n] = 2*dot(q_m, c_n) - xx[m] - xx[n]
// ---------------------------------------------------------------------------
__global__ __launch_bounds__(32) void knn_kernel(const float* __restrict__ x,
                                                 const float* __restrict__ xx,
                                                 int* __restrict__ knn) {
  __shared__ __align__(16) float qa[CH][16];   // qa[c][m] = x[b][c][q0+m]
  __shared__ float qxx[16];
  __shared__ __align__(16) float dtile[16][16];

  int wg = blockIdx.x;             // b * (N/16) + qtile
  int b  = wg / (NPTS / 16);
  int q0 = (wg - b * (NPTS / 16)) * 16;
  const float* xb  = x  + (size_t)b * CH * NPTS;
  const float* xxb = xx + (size_t)b * NPTS;
  int tid = threadIdx.x;

  for (int i = tid; i < CH * 16; i += 32) {
    int c = i >> 4, m = i & 15;
    qa[c][m] = xb[c * NPTS + q0 + m];
  }
  if (tid < 16) qxx[tid] = xxb[q0 + tid];
  __syncthreads();

  int lm = tid & 15;               // column / M lane within half-wave
  int hi = tid >> 4;               // half-wave select (K offset 0 or 2)

  // A fragments: invariant across all 128 column tiles -> registers (once).
  v2f afrag[16];
#pragma unroll
  for (int kc = 0; kc < 16; ++kc) {
    int k = kc * 4 + 2 * hi;
    afrag[kc].x = qa[k][lm];
    afrag[kc].y = qa[k + 1][lm];
  }

  float bestv[KNN];
  int   besti[KNN];
#pragma unroll
  for (int s = 0; s < KNN; ++s) { bestv[s] = -3.0e38f; besti[s] = 0; }

  const float* bbase = xb + lm;    // per-lane column base

  auto loadB = [&](v2f* bf, int j0) {
#pragma unroll
    for (int kc = 0; kc < 16; ++kc) {
      int k = kc * 4 + 2 * hi;
      bf[kc].x = bbase[k * NPTS + j0];
      bf[kc].y = bbase[(k + 1) * NPTS + j0];
    }
  };

  auto tile = [&](const v2f* bf, int j0) {
    v8f acc = {};
#pragma unroll
    for (int kc = 0; kc < 16; ++kc) acc = wmma_f32(afrag[kc], bf[kc], acc);
    float cxx = xxb[j0 + lm];
#pragma unroll
    for (int r = 0; r < 8; ++r) {
      int m = r + 8 * hi;
      dtile[m][lm] = 2.0f * acc[r] - qxx[m] - cxx;
    }
    __syncthreads();
    if (tid < 16) {                // lane `tid` owns query row q0+tid
#pragma unroll 4
      for (int j = 0; j < 16; ++j) {
        float v = dtile[tid][j];
        if (v > bestv[KNN - 1]) {  // beats current worst -> sorted insert
          float cv = v;
          int   ci = j0 + j;
#pragma unroll
          for (int s = 0; s < KNN; ++s) {
            bool  sw = cv > bestv[s];
            float tv = bestv[s];
            int   ti = besti[s];
            bestv[s] = sw ? cv : tv;
            besti[s] = sw ? ci : ti;
            cv = sw ? tv : cv;
            ci = sw ? ti : ci;
          }
        }
      }
    }
    __syncthreads();
  };

  v2f b0[16], b1[16];
  loadB(b0, 0);
  for (int jt = 0; jt < NPTS / 16; jt += 2) {
    if (jt + 2 < NPTS / 16)        // warm L0/GL2 two tiles ahead
      __builtin_prefetch(bbase + (jt + 2) * 16, 0, 3);
    loadB(b1, (jt + 1) * 16);      // loads overlap tile(b0) compute
    tile(b0, jt * 16);
    if (jt + 2 < NPTS / 16) loadB(b0, (jt + 2) * 16);
    tile(b1, (jt + 1) * 16);
  }

  if (tid < 16) {
    int* o = knn + ((size_t)b * NPTS + q0 + tid) * KNN;
#pragma unroll
    for (int s = 0; s < KNN; ++s) o[s] = besti[s];
  }
}

// ---------------------------------------------------------------------------
// Kernel 3: gather + edge conv (fp32 WMMA) + LeakyReLU + max over K.
// h[k][o] = W1 . feat_k + cterm[o],  cterm[o] = (W2-W1) . center + bias[o]
// One wave per point; A = 32x64 (rows 20..31 zero, masked in reduce).
// A fragments hoisted to registers and reused across all 4 N-tiles.
// ---------------------------------------------------------------------------
__global__ __launch_bounds__(32) void edge_kernel(const float* __restrict__ xt,
                                                  const int* __restrict__ knn,
                                                  const float* __restrict__ W,
                                                  const float* __restrict__ bias,
                                                  float* __restrict__ out) {
  __shared__ __align__(16) float fa[32][FAP];   // gathered features (A)
  __shared__ __align__(16) float wt[CH][OUTC];  // wt[c][o] = W[o][c] (B=W1^T)
  __shared__ float ctr[CH];
  __shared__ float cterm[OUTC];

  int gid = blockIdx.x;            // b * N + n
  int b = gid / NPTS;
  int n = gid - b * NPTS;
  int tid = threadIdx.x;

  // W1^T into LDS via b128 reads of W rows
  for (int i = tid; i < OUTC * (CH / 4); i += 32) {
    int o = i >> 4, c4 = (i & 15) * 4;
    float4 w = *(const float4*)&W[o * (2 * CH) + c4];
    wt[c4 + 0][o] = w.x;
    wt[c4 + 1][o] = w.y;
    wt[c4 + 2][o] = w.z;
    wt[c4 + 3][o] = w.w;
  }
  const float* xc = xt + (size_t)gid * CH;
  for (int c = tid; c < CH; c += 32) ctr[c] = xc[c];
  __syncthreads();

  // gather 20 neighbor rows (b128), zero-pad rows 20..31
  const int* id = knn + (size_t)gid * KNN;
  for (int i = tid; i < 32 * (CH / 4); i += 32) {
    int k = i >> 4, c4 = (i & 15) * 4;
    float4 v = {0.f, 0.f, 0.f, 0.f};
    if (k < KNN) {
      int nb = id[k];
      v = *(const float4*)&xt[((size_t)b * NPTS + nb) * CH + c4];
    }
    *(float4*)&fa[k][c4] = v;
  }

  // center term: each lane computes 2 of 64 outputs (independent of fa)
#pragma unroll
  for (int t = 0; t < 2; ++t) {
    int o = tid + 32 * t;
    const float* wr = W + (size_t)o * (2 * CH);
    float s = bias[o];
#pragma unroll
    for (int c = 0; c < CH; ++c) s += ctr[c] * (wr[CH + c] - wr[c]);
    cterm[o] = s;
  }
  __syncthreads();

  int lm = tid & 15;
  int hi = tid >> 4;

  // A fragments for both M-tiles: registers, reused across 4 N-tiles
  v2f afr[2][16];
#pragma unroll
  for (int mt = 0; mt < 2; ++mt) {
#pragma unroll
    for (int kc = 0; kc < 16; ++kc) {
      int k = kc * 4 + 2 * hi;
      afr[mt][kc] = *(const v2f*)&fa[mt * 16 + lm][k];
    }
  }

  for (int nt = 0; nt < 4; ++nt) {            // output-channel tiles
    int o = nt * 16 + lm;
    float ct = cterm[o];
    // B fragments for this N-tile (shared by both M-tiles)
    v2f bfr[16];
#pragma unroll
    for (int kc = 0; kc < 16; ++kc) {
      int k = kc * 4 + 2 * hi;
      bfr[kc].x = wt[k][o];
      bfr[kc].y = wt[k + 1][o];
    }
    float vmax = -3.0e38f;
#pragma unroll
    for (int mt = 0; mt < 2; ++mt) {          // neighbor tiles (k 0..15,16..31)
      v8f acc = {};
#pragma unroll
      for (int kc = 0; kc < 16; ++kc)
        acc = wmma_f32(afr[mt][kc], bfr[kc], acc);
#pragma unroll
      for (int r = 0; r < 8; ++r) {
        int kk = mt * 16 + r + 8 * hi;        // neighbor index of this element
        float h = acc[r] + ct;
        h = h >= 0.f ? h : ALPHA * h;         // LeakyReLU
        if (kk < KNN) vmax = fmaxf(vmax, h);
      }
    }
    // combine the two half-wave row groups (k rows 0-7 vs 8-15 per tile)
    vmax = fmaxf(vmax, __shfl_xor(vmax, 16, 32));
    if (hi == 0) out[((size_t)b * OUTC + o) * NPTS + n] = vmax;
  }
}

// ---------------------------------------------------------------------------
extern "C" void kernel_launch(void* const* d_in, const int* in_sizes, int n_in,
                              void* d_out, int out_size, void* d_ws,
                              size_t ws_size, hipStream_t stream) {
  (void)in_sizes; (void)n_in; (void)out_size; (void)ws_size;
  const float* x    = (const float*)d_in[0];
  const float* W    = (const float*)d_in[1];
  const float* bias = (const float*)d_in[2];
  float* out = (float*)d_out;

  char* ws = (char*)d_ws;
  size_t xt_bytes = (size_t)BATCH * NPTS * CH * sizeof(float);   // 4 MB
  size_t xx_bytes = (size_t)BATCH * NPTS * sizeof(float);        // 64 KB
  float* xt  = (float*)ws;
  float* xx  = (float*)(ws + xt_bytes);
  int*   knn = (int*)(ws + xt_bytes + xx_bytes);                 // 1.31 MB

  prep_kernel<<<(BATCH * NPTS + 255) / 256, 256, 0, stream>>>(x, xt, xx);
  knn_kernel<<<BATCH * (NPTS / 16), 32, 0, stream>>>(x, xx, knn);
  edge_kernel<<<BATCH * NPTS, 32, 0, stream>>>(xt, knn, W, bias, out);
}